// MultiGNNEncoder_44959717655082
// MI455X (gfx1250) — compile-verified
//
#include <hip/hip_runtime.h>
#include <math.h>

typedef __attribute__((ext_vector_type(2))) float v2f;
typedef __attribute__((ext_vector_type(8))) float v8f;

#define CCH 128  // output channels

// ---- monotone uint encoding of float for atomicMax over signed floats ----
__device__ __forceinline__ unsigned fenc(float f) {
  unsigned u = __float_as_uint(f);
  return (u & 0x80000000u) ? ~u : (u | 0x80000000u);
}
__device__ __forceinline__ float fdec(unsigned k) {
  unsigned u = (k & 0x80000000u) ? (k & 0x7fffffffu) : ~k;
  return __uint_as_float(u);
}

// ---------------- WMMA fp32 GEMM: H[n, 128] = X[n, K] @ W[K, 128] ----------
// One block = 8 waves = 16-row tile; wave w owns columns [16w, 16w+16).
// LDS A tile padded to stride K+4 dwords: stride % 64 == 4 -> the 16 lanes of
// each half-wave hit 16 distinct banks (conflict-free), and the (k, k+1) pair
// stays contiguous + 8B aligned so ds_load_b64 merging is preserved.
template <int K>
__global__ __launch_bounds__(256) void gemm_wmma_k(const float* __restrict__ X,
                                                   const float* __restrict__ W,
                                                   float* __restrict__ H,
                                                   int nrows) {
  constexpr int KP = K + 4;
  __shared__ float As[16 * KP];
  const int row0 = blockIdx.x * 16;
  // Stage the 16xK A tile in LDS once.
  for (int i = threadIdx.x; i < 16 * K; i += 256) {
    int r = i / K, k = i - r * K;
    int row = row0 + r;
    As[r * KP + k] = (row < nrows) ? X[(size_t)row * K + k] : 0.0f;
  }
  __syncthreads();

  const int wave  = threadIdx.x >> 5;
  const int lane  = threadIdx.x & 31;
  const int nlo   = lane & 15;          // M index for A, N index for B/C/D
  const int khalf = (lane >> 4) << 1;   // lanes 16-31 hold K+2,K+3
  const int col   = wave * 16 + nlo;

  v8f acc = {0.f, 0.f, 0.f, 0.f, 0.f, 0.f, 0.f, 0.f};

  for (int k0 = 0; k0 < K; k0 += 4) {
    v2f a, b;
    // A 16x4 layout (ISA 7.12.2): lane m holds (M=m, K=k0+khalf / +1)
    a.x = As[nlo * KP + k0 + khalf];
    a.y = As[nlo * KP + k0 + khalf + 1];
    // B 4x16 layout: lane n holds (K=k0+khalf / +1, N=n)
    b.x = W[(size_t)(k0 + khalf) * CCH + col];
    b.y = W[(size_t)(k0 + khalf + 1) * CCH + col];
    acc = __builtin_amdgcn_wmma_f32_16x16x4_f32(false, a, false, b,
                                                (short)0, acc, false, false);
  }

  // C/D layout: VGPR r -> M = r + 8*lane[4], N = col
  const int rbase = (lane >> 4) * 8;
  if (row0 + 16 <= nrows) {
    // full tile: branchless stores, immediate row offsets
    float* Hp = H + (size_t)(row0 + rbase) * CCH + col;
#pragma unroll
    for (int r = 0; r < 8; ++r) Hp[(size_t)r * CCH] = acc[r];
  } else {
    for (int r = 0; r < 8; ++r) {
      int row = row0 + rbase + r;
      if (row < nrows) H[(size_t)row * CCH + col] = acc[r];
    }
  }
}

// ---------------- a[row] = dot(H[row, :], att) ; one wave per row ----------
__global__ __launch_bounds__(256) void rowdot(const float* __restrict__ H,
                                              const float* __restrict__ att,
                                              float* __restrict__ a, int n) {
  int wave = threadIdx.x >> 5, lane = threadIdx.x & 31;
  int row = blockIdx.x * 8 + wave;
  if (row >= n) return;
  float4 h4 = reinterpret_cast<const float4*>(H + (size_t)row * CCH)[lane];
  float4 t4 = reinterpret_cast<const float4*>(att)[lane];
  float v = h4.x * t4.x + h4.y * t4.y + h4.z * t4.z + h4.w * t4.w;
  for (int m = 16; m >= 1; m >>= 1) v += __shfl_xor(v, m, 32);
  if (lane == 0) a[row] = v;
}

// ---------------- init: zero out accumulator, denom; mkey = encoded -inf ---
__global__ void init_k(float* __restrict__ out, float* __restrict__ denom,
                       unsigned* __restrict__ mkey, int nout, int ndst) {
  int i = blockIdx.x * blockDim.x + threadIdx.x;
  if (i < nout) out[i] = 0.0f;
  if (i < ndst) { denom[i] = 0.0f; mkey[i] = 0u; }  // key 0 < any real key
}

// ---------------- pass 1: e = leaky_relu(a_s+a_d); segment max via u32 max -
__global__ void edge_logit_max(const int* __restrict__ si, const int* __restrict__ di,
                               const float* __restrict__ a_src, const float* __restrict__ a_dst,
                               float* __restrict__ ebuf, unsigned* __restrict__ mkey, int E) {
  int i = blockIdx.x * blockDim.x + threadIdx.x;
  if (i >= E) return;
  float x = a_src[si[i]] + a_dst[di[i]];
  float e = x > 0.0f ? x : 0.2f * x;
  ebuf[i] = e;
  atomicMax(&mkey[di[i]], fenc(e));
}

// ---------------- pass 2: p = exp(e - m[d]); denom[d] += p ------------------
__global__ void edge_exp_sum(const int* __restrict__ di, const unsigned* __restrict__ mkey,
                             float* __restrict__ ebuf, float* __restrict__ denom, int E) {
  int i = blockIdx.x * blockDim.x + threadIdx.x;
  if (i >= E) return;
  int d = di[i];
  float p = __expf(ebuf[i] - fdec(mkey[d]));
  ebuf[i] = p;  // reuse buffer for p
  atomicAdd(&denom[d], p);
}

// ---------------- pass 3: out[d, :] += alpha * h_src[s, :]; wave per edge ---
__global__ __launch_bounds__(256) void edge_scatter(const int* __restrict__ si,
                                                    const int* __restrict__ di,
                                                    const float* __restrict__ H,
                                                    const float* __restrict__ pbuf,
                                                    const float* __restrict__ denom,
                                                    float* __restrict__ out, int E) {
  int wave = threadIdx.x >> 5, lane = threadIdx.x & 31;
  long long e = (long long)blockIdx.x * 8 + wave;
  if (e >= E) return;
  int s = si[e], d = di[e];
  float alpha = pbuf[e] / (denom[d] + 1e-16f);
  float4 h4 = reinterpret_cast<const float4*>(H + (size_t)s * CCH)[lane];  // coalesced 512B/wave
  float* o = out + (size_t)d * CCH + lane * 4;
  atomicAdd(o + 0, h4.x * alpha);
  atomicAdd(o + 1, h4.y * alpha);
  atomicAdd(o + 2, h4.z * alpha);
  atomicAdd(o + 3, h4.w * alpha);
}

// ---------------- finalize: out = relu(out + bias) --------------------------
__global__ void bias_relu(float* __restrict__ out, const float* __restrict__ bias, int n) {
  int i = blockIdx.x * blockDim.x + threadIdx.x;
  if (i >= n) return;
  float v = out[i] + bias[i & (CCH - 1)];
  out[i] = v > 0.0f ? v : 0.0f;
}

extern "C" void kernel_launch(void* const* d_in, const int* in_sizes, int n_in,
                              void* d_out, int out_size, void* d_ws, size_t ws_size,
                              hipStream_t stream) {
  const float* x_src   = (const float*)d_in[0];
  const float* x_dst   = (const float*)d_in[1];
  const int*   src_idx = (const int*)d_in[2];
  const int*   dst_idx = (const int*)d_in[3];
  const float* W_src   = (const float*)d_in[4];
  const float* W_dst   = (const float*)d_in[5];
  const float* att_src = (const float*)d_in[6];
  const float* att_dst = (const float*)d_in[7];
  const float* bias    = (const float*)d_in[8];
  float* out = (float*)d_out;

  const int D_SRC = 256, D_DST = 128;
  const int Nsrc = in_sizes[0] / D_SRC;
  const int Ndst = in_sizes[1] / D_DST;
  const int E    = in_sizes[2];

  // workspace layout (bytes)
  char* ws = (char*)d_ws;
  float*    h_src = (float*)ws;                               // Nsrc*128
  float*    h_dst = (float*)(ws + (size_t)Nsrc * CCH * 4);    // Ndst*128
  float*    a_src = (float*)((char*)h_dst + (size_t)Ndst * CCH * 4);
  float*    a_dst = a_src + Nsrc;
  float*    denom = a_dst + Ndst;
  unsigned* mkey  = (unsigned*)(denom + Ndst);
  float*    ebuf  = (float*)(mkey + Ndst);                    // E floats

  // 1) dense projections via fp32 WMMA
  gemm_wmma_k<256><<<(Nsrc + 15) / 16, 256, 0, stream>>>(x_src, W_src, h_src, Nsrc);
  gemm_wmma_k<128><<<(Ndst + 15) / 16, 256, 0, stream>>>(x_dst, W_dst, h_dst, Ndst);

  // 2) attention scalars
  rowdot<<<(Nsrc + 7) / 8, 256, 0, stream>>>(h_src, att_src, a_src, Nsrc);
  rowdot<<<(Ndst + 7) / 8, 256, 0, stream>>>(h_dst, att_dst, a_dst, Ndst);

  // 3) init accumulators
  int nout = Ndst * CCH;
  init_k<<<(nout + 255) / 256, 256, 0, stream>>>(out, denom, mkey, nout, Ndst);

  // 4) segment softmax (3 edge passes) + scatter aggregate
  edge_logit_max<<<(E + 255) / 256, 256, 0, stream>>>(src_idx, dst_idx, a_src, a_dst, ebuf, mkey, E);
  edge_exp_sum<<<(E + 255) / 256, 256, 0, stream>>>(dst_idx, mkey, ebuf, denom, E);
  edge_scatter<<<(E + 7) / 8, 256, 0, stream>>>(src_idx, dst_idx, h_src, ebuf, denom, out, E);

  // 5) bias + relu
  bias_relu<<<(nout + 255) / 256, 256, 0, stream>>>(out, bias, nout);
}